// AttnDecoderRNN_55190329754055
// MI455X (gfx1250) — compile-verified
//
#include <hip/hip_runtime.h>
#include <math.h>

typedef __attribute__((ext_vector_type(2))) float v2f;
typedef __attribute__((ext_vector_type(8))) float v8f;

#define NB 64      // batch
#define HH 1024
#define HH2 2048
#define VV 32000
#define LL 64
#define PADIDX 2

// ---------------------------------------------------------------------------
// Skinny GEMM:  Out[b, n] = sum_k X[b,k] * W[n,k]  (+bias),  b = 0..63.
// Two optional (X, W) passes accumulate into the same tile (LSTM x@Wih + h@Whh).
// One wave owns a 16-wide N tile and the FULL batch (4 M-tiles of 16), so each
// weight element is streamed from HBM exactly once.
// Inner loop is software-pipelined: each stage loads a 16-k group (4 B float2 +
// 16 A float2, all independent -> deep loadcnt pipelining) while the WMMAs of
// the previous group execute. K must be a multiple of 32 (2048/3072/5120 all are).
// TRANSB=false: W is (N, K) row-major  -> per-lane contiguous float2 loads
// TRANSB=true : W is (K, N) row-major  (Ua_w case) -> strided scalar loads
// ---------------------------------------------------------------------------

#define LOADG(KB, bb, A0, A1, A2, A3)                                          \
  _Pragma("unroll")                                                            \
  for (int j = 0; j < 4; ++j) {                                                \
    const int kk = (KB) + 4 * j;                                               \
    if (TRANSB) {                                                              \
      const float* wp = W + (size_t)(kk + koff) * ldw + n;                     \
      bb[j].x = wp[0];                                                         \
      bb[j].y = wp[ldw];                                                       \
    } else {                                                                   \
      bb[j] = *(const v2f*)(wr + kk);                                          \
    }                                                                          \
    A0[j] = *(const v2f*)(xr0 + kk);                                           \
    A1[j] = *(const v2f*)(xr1 + kk);                                           \
    A2[j] = *(const v2f*)(xr2 + kk);                                           \
    A3[j] = *(const v2f*)(xr3 + kk);                                           \
  }

#define MMAG(bb, A0, A1, A2, A3)                                               \
  _Pragma("unroll")                                                            \
  for (int j = 0; j < 4; ++j) {                                                \
    acc0 = __builtin_amdgcn_wmma_f32_16x16x4_f32(false, A0[j], false, bb[j], (short)0, acc0, false, false); \
    acc1 = __builtin_amdgcn_wmma_f32_16x16x4_f32(false, A1[j], false, bb[j], (short)0, acc1, false, false); \
    acc2 = __builtin_amdgcn_wmma_f32_16x16x4_f32(false, A2[j], false, bb[j], (short)0, acc2, false, false); \
    acc3 = __builtin_amdgcn_wmma_f32_16x16x4_f32(false, A3[j], false, bb[j], (short)0, acc3, false, false); \
  }

template <bool TRANSB>
__global__ __launch_bounds__(128) void gemm64_wmma(
    const float* __restrict__ X0, const float* __restrict__ W0,
    const float* __restrict__ X1, const float* __restrict__ W1,
    const float* __restrict__ bias0, const float* __restrict__ bias1,
    float* __restrict__ Out, int N, int K, int ldx, int ldw, int ldo)
{
  const int lane = threadIdx.x & 31;
  const int nt   = blockIdx.x * (blockDim.x >> 5) + (threadIdx.x >> 5);
  const int n0   = nt * 16;
  if (n0 >= N) return;                 // wave-uniform: EXEC stays all-ones for WMMA
  const int nl   = lane & 15;          // A: M row; B/D: N col
  const int koff = (lane >> 4) << 1;   // K sub-offset per lane half (0 or 2)
  const int n    = n0 + nl;

  v8f acc0 = {}, acc1 = {}, acc2 = {}, acc3 = {};

  for (int pass = 0; pass < 2; ++pass) {
    const float* __restrict__ X = pass ? X1 : X0;
    const float* __restrict__ W = pass ? W1 : W0;
    if (!X) continue;
    const float* xr0 = X + (0 * 16 + nl) * ldx + koff;
    const float* xr1 = X + (1 * 16 + nl) * ldx + koff;
    const float* xr2 = X + (2 * 16 + nl) * ldx + koff;
    const float* xr3 = X + (3 * 16 + nl) * ldx + koff;
    const float* wr  = W + (size_t)n * ldw + koff;   // non-TRANS base

    v2f bA[4], a0A[4], a1A[4], a2A[4], a3A[4];
    v2f bB[4], a0B[4], a1B[4], a2B[4], a3B[4];

    LOADG(0, bA, a0A, a1A, a2A, a3A);
    int k0 = 0;
    for (; k0 + 32 < K; k0 += 32) {
      LOADG(k0 + 16, bB, a0B, a1B, a2B, a3B);
      MMAG(bA, a0A, a1A, a2A, a3A);
      LOADG(k0 + 32, bA, a0A, a1A, a2A, a3A);
      MMAG(bB, a0B, a1B, a2B, a3B);
    }
    LOADG(k0 + 16, bB, a0B, a1B, a2B, a3B);
    MMAG(bA, a0A, a1A, a2A, a3A);
    MMAG(bB, a0B, a1B, a2B, a3B);
  }

  float bsum = 0.f;
  if (bias0) bsum += bias0[n];
  if (bias1) bsum += bias1[n];
  // D layout: col = lane&15 ; row = 8*(lane>>4) + vgpr
  const int rbase = (lane >> 4) * 8;
  float* op = Out + n0 + nl;
#pragma unroll
  for (int i = 0; i < 8; ++i) {
    op[(0 * 16 + rbase + i) * (size_t)ldo] = acc0[i] + bsum;
    op[(1 * 16 + rbase + i) * (size_t)ldo] = acc1[i] + bsum;
    op[(2 * 16 + rbase + i) * (size_t)ldo] = acc2[i] + bsum;
    op[(3 * 16 + rbase + i) * (size_t)ldo] = acc3[i] + bsum;
  }
}

// ---------------------------------------------------------------------------
__global__ void gather_emb(const int* __restrict__ wb, const float* __restrict__ emb,
                           float* __restrict__ out)
{
  const int b = blockIdx.x;
  const size_t row = (size_t)wb[b];
  const float4* src = (const float4*)(emb + row * HH);
  float4* dst = (float4*)(out + (size_t)b * HH);
  for (int i = threadIdx.x; i < HH / 4; i += blockDim.x) dst[i] = src[i];
}

// scores -> mask -> softmax(L) -> context, fused per batch row
__global__ __launch_bounds__(256) void attn_kernel(
    const float* __restrict__ u,    // (B, 2048) = h3 @ Ua_w
    const float* __restrict__ enc,  // (L, B, 2048)
    const int* __restrict__ inp,    // (L, B)
    float* __restrict__ attn_out,   // (B, L) -> d_out region
    float* __restrict__ ctx)        // (B, 2048)
{
  const int b = blockIdx.x;
  const int tid = threadIdx.x, lane = tid & 31, wave = tid >> 5;
  __shared__ float s[LL];
  __shared__ float wgt[LL];

  const float* ub = u + (size_t)b * HH2;
  for (int li = 0; li < 8; ++li) {
    const int l = wave * 8 + li;
    const float* e = enc + (size_t)l * NB * HH2 + (size_t)b * HH2;
    float p = 0.f;
    for (int k = lane; k < HH2; k += 32) p += ub[k] * e[k];
    for (int off = 16; off; off >>= 1) p += __shfl_xor(p, off, 32);
    if (lane == 0) s[l] = (inp[l * NB + b] == PADIDX) ? -__builtin_inff() : p;
  }
  __syncthreads();

  if (wave == 0) {
    float a0 = s[lane], a1 = s[lane + 32];
    float mx = fmaxf(a0, a1);
    for (int off = 16; off; off >>= 1) mx = fmaxf(mx, __shfl_xor(mx, off, 32));
    float e0 = expf(a0 - mx), e1 = expf(a1 - mx);
    float sm = e0 + e1;
    for (int off = 16; off; off >>= 1) sm += __shfl_xor(sm, off, 32);
    const float inv = 1.f / sm;
    wgt[lane] = e0 * inv;
    wgt[lane + 32] = e1 * inv;
    attn_out[b * LL + lane] = e0 * inv;
    attn_out[b * LL + lane + 32] = e1 * inv;
  }
  __syncthreads();

  for (int d = tid; d < HH2; d += 256) {
    float acc = 0.f;
    for (int l = 0; l < LL; ++l)
      acc += wgt[l] * enc[(size_t)l * NB * HH2 + (size_t)b * HH2 + d];
    ctx[(size_t)b * HH2 + d] = acc;
  }
}

__global__ void concat3(const float* __restrict__ embd, const float* __restrict__ ctx,
                        float* __restrict__ cat3)
{
  const int idx = blockIdx.x * 256 + threadIdx.x;
  if (idx >= NB * 3 * HH) return;
  const int b = idx / (3 * HH), k = idx % (3 * HH);
  cat3[idx] = (k < HH) ? embd[b * HH + k] : ctx[b * HH2 + (k - HH)];
}

__global__ void concat5(const float* __restrict__ h4, const float* __restrict__ ctx,
                        const float* __restrict__ embd, float* __restrict__ cat5)
{
  const int idx = blockIdx.x * 256 + threadIdx.x;
  if (idx >= NB * 5 * HH) return;
  const int b = idx / (5 * HH), k = idx % (5 * HH);
  float v;
  if (k < HH2)            v = h4[b * HH2 + k];
  else if (k < 2 * HH2)   v = ctx[b * HH2 + (k - HH2)];
  else                    v = embd[b * HH + (k - 2 * HH2)];
  cat5[idx] = v;
}

__device__ __forceinline__ float sigf(float x) { return 1.f / (1.f + expf(-x)); }

// gate order i, f, g, o (PyTorch)
__global__ void lstm_pointwise(const float* __restrict__ g, const float* __restrict__ cin,
                               float* __restrict__ hout, float* __restrict__ cout,
                               const float* __restrict__ addsrc, float* __restrict__ sumout)
{
  const int idx = blockIdx.x * 256 + threadIdx.x;   // B*2048
  if (idx >= NB * HH2) return;
  const int b = idx >> 11, j = idx & (HH2 - 1);
  const float* gb = g + (size_t)b * 4 * HH2;
  const float i  = sigf(gb[j]);
  const float f  = sigf(gb[HH2 + j]);
  const float gg = tanhf(gb[2 * HH2 + j]);
  const float o  = sigf(gb[3 * HH2 + j]);
  const float c2 = f * cin[idx] + i * gg;
  const float h2 = o * tanhf(c2);
  cout[idx] = c2;
  hout[idx] = h2;
  if (sumout) sumout[idx] = h2 + addsrc[idx];       // feeds cell3 input (h2 + h3)
}

__global__ __launch_bounds__(256) void softmaxV(float* __restrict__ logits)
{
  const int b = blockIdx.x, t = threadIdx.x;
  float* row = logits + (size_t)b * VV;
  __shared__ float red[256];
  float mx = -__builtin_inff();
  for (int i = t; i < VV; i += 256) mx = fmaxf(mx, row[i]);
  red[t] = mx; __syncthreads();
  for (int s = 128; s; s >>= 1) { if (t < s) red[t] = fmaxf(red[t], red[t + s]); __syncthreads(); }
  mx = red[0]; __syncthreads();
  float sum = 0.f;
  for (int i = t; i < VV; i += 256) { const float e = expf(row[i] - mx); row[i] = e; sum += e; }
  red[t] = sum; __syncthreads();
  for (int s = 128; s; s >>= 1) { if (t < s) red[t] += red[t + s]; __syncthreads(); }
  const float inv = 1.f / red[0];
  for (int i = t; i < VV; i += 256) row[i] *= inv;
}

// ---------------------------------------------------------------------------
extern "C" void kernel_launch(void* const* d_in, const int* in_sizes, int n_in,
                              void* d_out, int out_size, void* d_ws, size_t ws_size,
                              hipStream_t stream)
{
  const int*   word_batch = (const int*)d_in[0];
  const float* enc        = (const float*)d_in[1];
  const int*   input_b    = (const int*)d_in[2];
  const float* h_in       = (const float*)d_in[3];   // (4,1,B,2H)
  const float* c_in       = (const float*)d_in[4];
  const float* emb        = (const float*)d_in[5];
  const float* Ua_w       = (const float*)d_in[6];   // (2H,2H)
  const float* comb_w     = (const float*)d_in[7];   // (2H,3H)
  const float* comb_b     = (const float*)d_in[8];
  const float* out_w      = (const float*)d_in[9];   // (V,5H)
  const float* out_b      = (const float*)d_in[10];
  const float *wih[4], *whh[4], *bih[4], *bhh[4];
  for (int i = 0; i < 4; ++i) {
    wih[i] = (const float*)d_in[11 + 4 * i];
    whh[i] = (const float*)d_in[12 + 4 * i];
    bih[i] = (const float*)d_in[13 + 4 * i];
    bhh[i] = (const float*)d_in[14 + 4 * i];
  }

  const int SZ = NB * HH2;                 // 131072 floats per (B,2H) slab
  float* ws   = (float*)d_ws;
  float* embd = ws;                        //  65536
  float* u    = embd + NB * HH;            // 131072
  float* ctx  = u    + SZ;                 // 131072
  float* cat3 = ctx  + SZ;                 // 196608
  float* x0   = cat3 + NB * 3 * HH;        // 131072
  float* gbuf = x0   + SZ;                 // 524288
  float* x23  = gbuf + NB * 4 * HH2;       // 131072
  float* cat5 = x23  + SZ;                 // 327680

  float* outp = (float*)d_out;
  float* hs   = outp + (size_t)NB * VV;    // (4,1,B,2H)
  float* cs   = hs + 4 * SZ;
  float* attn = cs + 4 * SZ;               // (B,L)

  // 1) embedding gather
  gather_emb<<<NB, 256, 0, stream>>>(word_batch, emb, embd);

  // 2) u = h3 @ Ua_w   (algebraic rewrite of the enc@Ua_w.T GEMM: 64x smaller)
  gemm64_wmma<true><<<32, 128, 0, stream>>>(h_in + 3 * SZ, Ua_w, nullptr, nullptr,
                                            nullptr, nullptr, u, HH2, HH2, HH2, HH2, HH2);

  // 3) scores + mask + softmax(L) + context (fused)
  attn_kernel<<<NB, 256, 0, stream>>>(u, enc, input_b, attn, ctx);

  // 4) lstm_in = [embedded|context] @ comb_w.T + comb_b
  concat3<<<(NB * 3 * HH + 255) / 256, 256, 0, stream>>>(embd, ctx, cat3);
  gemm64_wmma<false><<<32, 128, 0, stream>>>(cat3, comb_w, nullptr, nullptr,
                                             comb_b, nullptr, x0, HH2, 3 * HH, 3 * HH, 3 * HH, HH2);

  // 5) four LSTM cells: fused (x@Wih + h@Whh + biases) GEMM, then pointwise
  const float* xin[4] = { x0, hs, hs + SZ, x23 };
  for (int i = 0; i < 4; ++i) {
    gemm64_wmma<false><<<128, 128, 0, stream>>>(xin[i], wih[i], h_in + (size_t)i * SZ, whh[i],
                                                bih[i], bhh[i], gbuf, 4 * HH2, HH2, HH2, HH2, 4 * HH2);
    const float* addsrc = (i == 2) ? (hs + SZ) : nullptr;   // h2 (+ h3) -> cell3 input
    float* sumout       = (i == 2) ? x23 : nullptr;
    lstm_pointwise<<<(SZ + 255) / 256, 256, 0, stream>>>(gbuf, c_in + (size_t)i * SZ,
                                                         hs + (size_t)i * SZ, cs + (size_t)i * SZ,
                                                         addsrc, sumout);
  }

  // 6) logits = [h4|context|embedded] @ out_w.T + out_b ; softmax over V (in place)
  concat5<<<(NB * 5 * HH + 255) / 256, 256, 0, stream>>>(hs + 3 * SZ, ctx, embd, cat5);
  gemm64_wmma<false><<<500, 128, 0, stream>>>(cat5, out_w, nullptr, nullptr,
                                              out_b, nullptr, outp, VV, 5 * HH, 5 * HH, 5 * HH, VV);
  softmaxV<<<NB, 256, 0, stream>>>(outp);
}